// QActivation_64115271795097
// MI455X (gfx1250) — compile-verified
//
#include <hip/hip_runtime.h>

typedef float vf4 __attribute__((ext_vector_type(4)));
typedef float v2f __attribute__((ext_vector_type(2)));
typedef float v8f __attribute__((ext_vector_type(8)));

#define BETA_F 0.5f
#define B_DIM 256
#define D_DIM 512

// ---------------------------------------------------------------------------
// Bulk streaming kernel: out = 0.5 * in, 16B per lane, non-temporal both ways
// (1.07 GB total traffic; bandwidth-bound, ~46us floor at 23.3 TB/s).
// ---------------------------------------------------------------------------
__global__ void scale_stream_kernel(const vf4* __restrict__ in,
                                    vf4* __restrict__ out, int n4) {
    int i = blockIdx.x * blockDim.x + threadIdx.x;
    if (i < n4) {
        vf4 v = __builtin_nontemporal_load(in + i);
        v *= BETA_F;
        __builtin_nontemporal_store(v, out + i);
    }
}

// ---------------------------------------------------------------------------
// Diagonal softmax + fixup. One wave32 per batch (8 waves / 256-thread block).
// Each lane owns 16 diagonal elements (D=512 = 32 lanes * 16).
// Sum-of-exp reduction uses V_WMMA_F32_16X16X4_F32 with B = ones:
//   A layout (16x4 f32): lane L (0..15) VGPR0 -> A[L,0]; lane L+16 VGPR0 -> A[L,2];
//   VGPR1 (K=1,K=3) set to 0.  With B=1: D[m,n] = partial_m + partial_{m+16}.
//   Sum of 8 C/D VGPRs per lane + shfl_xor(16) = full 32-lane sum everywhere.
// ---------------------------------------------------------------------------
__global__ void diag_softmax_kernel(const float* __restrict__ x_real,
                                    float* __restrict__ out_r) {
    const int D = D_DIM;
    const int wave = threadIdx.x >> 5;
    const int lane = threadIdx.x & 31;
    const int b = blockIdx.x * 8 + wave;

    const float* base = x_real + (size_t)b * D * D;
    float* obase = out_r + (size_t)b * D * D;

    // Gather 16 diagonal values per lane, track local max.
    float d[16];
    float m = -3.402823466e38f;
#pragma unroll
    for (int j = 0; j < 16; ++j) {
        int i = lane + 32 * j;
        d[j] = base[(size_t)i * (D + 1)];
        m = fmaxf(m, d[j]);
    }

    // Wave max reduction (wave32).
#pragma unroll
    for (int off = 16; off >= 1; off >>= 1)
        m = fmaxf(m, __shfl_xor(m, off, 32));

    // Per-lane partial sum of exp.
    float p = 0.0f;
#pragma unroll
    for (int j = 0; j < 16; ++j)
        p += __expf(d[j] - m);

    // Wave sum reduction via WMMA f32 16x16x4 with all-ones B.
    v2f a;  a.x = p;    a.y = 0.0f;
    v2f bo; bo.x = 1.0f; bo.y = 1.0f;
    v8f c = {};
    c = __builtin_amdgcn_wmma_f32_16x16x4_f32(
        /*neg_a=*/false, a, /*neg_b=*/false, bo,
        /*c_mod=*/(short)0, c, /*reuse_a=*/false, /*reuse_b=*/false);
    float s = c[0] + c[1] + c[2] + c[3] + c[4] + c[5] + c[6] + c[7];
    s += __shfl_xor(s, 16, 32);

    const float inv = 1.0f / s;

    // Final diagonal values: 0.5*x + 0.5*softmax  (overwrites bulk-scaled diag).
#pragma unroll
    for (int j = 0; j < 16; ++j) {
        int i = lane + 32 * j;
        obase[(size_t)i * (D + 1)] =
            BETA_F * d[j] + (1.0f - BETA_F) * (__expf(d[j] - m) * inv);
    }
}

// ---------------------------------------------------------------------------
// Launch: two bulk streams (real, imag halves of d_out), then diagonal fixup.
// Stream ordering makes the fixup land after the bulk writes.
// ---------------------------------------------------------------------------
extern "C" void kernel_launch(void* const* d_in, const int* in_sizes, int n_in,
                              void* d_out, int out_size, void* d_ws, size_t ws_size,
                              hipStream_t stream) {
    (void)in_sizes; (void)n_in; (void)out_size; (void)d_ws; (void)ws_size;

    const float* x_real = (const float*)d_in[0];
    const float* x_imag = (const float*)d_in[1];
    float* out = (float*)d_out;

    const size_t half = (size_t)B_DIM * D_DIM * D_DIM;   // 67,108,864 floats
    const int n4 = (int)(half / 4);                      // 16,777,216 float4s

    dim3 blk(256);
    dim3 grd((unsigned)(n4 / 256));                      // 65,536 blocks

    scale_stream_kernel<<<grd, blk, 0, stream>>>(
        (const vf4*)x_real, (vf4*)out, n4);
    scale_stream_kernel<<<grd, blk, 0, stream>>>(
        (const vf4*)x_imag, (vf4*)(out + half), n4);

    diag_softmax_kernel<<<dim3(B_DIM / 8), dim3(256), 0, stream>>>(
        x_real, out);
}